// mlp_72816875536992
// MI455X (gfx1250) — compile-verified
//
#include <hip/hip_runtime.h>

// ---------------------------------------------------------------------------
// Fused 3-layer MLP (32 -> 32 -> 32 -> 3), fp32, for MI455X (gfx1250).
// Memory-bound (AI ~32 flop/B): stream x at HBM rate, compute with the exact
// fp32 WMMA  V_WMMA_F32_16X16X4_F32  (K=4 per issue, 8 issues per layer-K=32).
// One 16-row tile per wave32; weights live in per-wave VGPR B-fragments.
// ---------------------------------------------------------------------------

typedef float v2f __attribute__((ext_vector_type(2)));
typedef float v4f __attribute__((ext_vector_type(4)));
typedef float v8f __attribute__((ext_vector_type(8)));

#define RSTR 36               // LDS row stride in floats (conflict-free, 16B aligned)
#define TILE_ROWS 16
#define WAVES_PER_BLOCK 8

static __device__ __forceinline__ v8f wmma_f32(v2f a, v2f b, v8f c) {
  // D = A(16x4,f32) * B(4x16,f32) + C(16x16,f32); exact fp32.
  return __builtin_amdgcn_wmma_f32_16x16x4_f32(
      /*neg_a=*/false, a, /*neg_b=*/false, b,
      /*c_mod=*/(short)0, c, /*reuse_a=*/false, /*reuse_b=*/false);
}

// B-fragment for one 4x16 chunk of W (row-major, ldW floats/row, ncols valid):
//   VGPR0: lanes 0-15 -> W[k0+0][c*16+n], lanes 16-31 -> W[k0+2][c*16+n]
//   VGPR1: lanes 0-15 -> W[k0+1][c*16+n], lanes 16-31 -> W[k0+3][c*16+n]
static __device__ __forceinline__ v2f load_bfrag(const float* W, int ldW,
                                                 int ncols, int k0, int ctile,
                                                 int lane) {
  int n = (lane & 15) + ctile * 16;
  int k = k0 + ((lane >> 4) << 1);
  v2f r;
  r.x = (n < ncols) ? W[(k + 0) * ldW + n] : 0.0f;
  r.y = (n < ncols) ? W[(k + 1) * ldW + n] : 0.0f;
  return r;
}

__global__ void __launch_bounds__(256)
mlp_fused_wmma_kernel(const float* __restrict__ x,
                      const float* __restrict__ W0, const float* __restrict__ b0,
                      const float* __restrict__ W1, const float* __restrict__ b1,
                      const float* __restrict__ W2, const float* __restrict__ b2,
                      float* __restrict__ out, int numTiles) {
  __shared__ float lds[WAVES_PER_BLOCK][2][TILE_ROWS * RSTR];

  const int lane = threadIdx.x & 31;
  const int wave = threadIdx.x >> 5;
  const int nIdx = lane & 15;
  const int kHi  = (lane >> 4) << 1;   // 0 or 2
  const int mHi  = (lane >> 4) << 3;   // 0 or 8 (C/D row offset)

  float* buf0 = &lds[wave][0][0];
  float* buf1 = &lds[wave][1][0];

  // ---- Preload weight B-fragments into VGPRs (loop-invariant) -------------
  v2f w0f[8][2], w1f[8][2], w2f[8];
#pragma unroll
  for (int k = 0; k < 8; ++k) {
    w0f[k][0] = load_bfrag(W0, 32, 32, k * 4, 0, lane);
    w0f[k][1] = load_bfrag(W0, 32, 32, k * 4, 1, lane);
    w1f[k][0] = load_bfrag(W1, 32, 32, k * 4, 0, lane);
    w1f[k][1] = load_bfrag(W1, 32, 32, k * 4, 1, lane);
    w2f[k]    = load_bfrag(W2,  3,  3, k * 4, 0, lane);
  }
  // Per-lane biases in C/D layout (bias depends only on output column = nIdx)
  const float bias0a = b0[nIdx], bias0b = b0[16 + nIdx];
  const float bias1a = b1[nIdx], bias1b = b1[16 + nIdx];
  const float bias2  = (nIdx < 3) ? b2[nIdx] : 0.0f;

  const int waveGlobal = blockIdx.x * WAVES_PER_BLOCK + wave;
  const int waveStride = gridDim.x * WAVES_PER_BLOCK;

  for (int t = waveGlobal; t < numTiles; t += waveStride) {
    // ---- Stage 16x32 fp32 x-tile into LDS, fully coalesced (4 x b128) ----
    const float* xt = x + (size_t)t * (TILE_ROWS * 32);
#pragma unroll
    for (int j = 0; j < 4; ++j) {
      int idx = j * 128 + lane * 4;          // linear float index in tile
      v4f v = *(const v4f*)(xt + idx);
      int r = idx >> 5, c = idx & 31;
      *(v4f*)(buf0 + r * RSTR + c) = v;
    }
    asm volatile("s_wait_dscnt 0" ::: "memory");

    // ---- A-fragments of x: lane n holds row n, K from VGPR index ---------
    v2f xa[8];
#pragma unroll
    for (int k = 0; k < 8; ++k)
      xa[k] = *(const v2f*)(buf0 + nIdx * RSTR + k * 4 + kHi);

    // ---- Layer 0: h0 = relu(x @ W0 + b0)  (two 16x16 col-tiles) ----------
    v8f acc0 = {}, acc1 = {};
#pragma unroll
    for (int k = 0; k < 8; ++k) {
      acc0 = wmma_f32(xa[k], w0f[k][0], acc0);
      acc1 = wmma_f32(xa[k], w0f[k][1], acc1);
    }
#pragma unroll
    for (int g = 0; g < 8; ++g) {
      acc0[g] = fmaxf(acc0[g] + bias0a, 0.0f);
      acc1[g] = fmaxf(acc1[g] + bias0b, 0.0f);
    }

    // ---- Transpose h0 (C/D layout -> A layout) via LDS -------------------
#pragma unroll
    for (int g = 0; g < 8; ++g) {
      int row = g + mHi;
      buf1[row * RSTR + nIdx]      = acc0[g];
      buf1[row * RSTR + 16 + nIdx] = acc1[g];
    }
    asm volatile("s_wait_dscnt 0" ::: "memory");
    v2f ha[8];
#pragma unroll
    for (int k = 0; k < 8; ++k)
      ha[k] = *(const v2f*)(buf1 + nIdx * RSTR + k * 4 + kHi);

    // ---- Layer 1: h1 = relu(h0 @ W1 + b1) --------------------------------
    acc0 = (v8f){}; acc1 = (v8f){};
#pragma unroll
    for (int k = 0; k < 8; ++k) {
      acc0 = wmma_f32(ha[k], w1f[k][0], acc0);
      acc1 = wmma_f32(ha[k], w1f[k][1], acc1);
    }
#pragma unroll
    for (int g = 0; g < 8; ++g) {
      acc0[g] = fmaxf(acc0[g] + bias1a, 0.0f);
      acc1[g] = fmaxf(acc1[g] + bias1b, 0.0f);
    }

    // ---- Transpose h1 via LDS --------------------------------------------
#pragma unroll
    for (int g = 0; g < 8; ++g) {
      int row = g + mHi;
      buf0[row * RSTR + nIdx]      = acc0[g];
      buf0[row * RSTR + 16 + nIdx] = acc1[g];
    }
    asm volatile("s_wait_dscnt 0" ::: "memory");
    v2f h2[8];
#pragma unroll
    for (int k = 0; k < 8; ++k)
      h2[k] = *(const v2f*)(buf0 + nIdx * RSTR + k * 4 + kHi);

    // ---- Layer 2: out = h1 @ W2 + b2  (cols >= 3 are zero-padded) --------
    v8f acc2 = {};
#pragma unroll
    for (int k = 0; k < 8; ++k)
      acc2 = wmma_f32(h2[k], w2f[k], acc2);
#pragma unroll
    for (int g = 0; g < 8; ++g)
      acc2[g] += bias2;

    // ---- Transpose 16x3 result via LDS, then one coalesced 192B store ----
#pragma unroll
    for (int g = 0; g < 8; ++g)
      buf1[(g + mHi) * RSTR + nIdx] = acc2[g];
    asm volatile("s_wait_dscnt 0" ::: "memory");

    float* ot = out + (size_t)t * (TILE_ROWS * 3);
    {
      int i = lane;                               // floats 0..31 of 48
      ot[i] = buf1[(i / 3) * RSTR + (i % 3)];
      if (lane < 16) {
        int j2 = 32 + lane;                       // floats 32..47
        ot[j2] = buf1[(j2 / 3) * RSTR + (j2 % 3)];
      }
    }
  }
}

extern "C" void kernel_launch(void* const* d_in, const int* in_sizes, int n_in,
                              void* d_out, int out_size, void* d_ws, size_t ws_size,
                              hipStream_t stream) {
  // setup_inputs() dict order: x, W0, b0, W1, b1, W2, b2
  const float* x  = (const float*)d_in[0];
  const float* W0 = (const float*)d_in[1];
  const float* b0 = (const float*)d_in[2];
  const float* W1 = (const float*)d_in[3];
  const float* b1 = (const float*)d_in[4];
  const float* W2 = (const float*)d_in[5];
  const float* b2 = (const float*)d_in[6];
  float* out = (float*)d_out;

  const int N = in_sizes[0] / 32;      // rows of x
  const int numTiles = N / TILE_ROWS;  // 262144 for N = 4,194,304

  dim3 grid(2048), block(256);         // 16K waves, grid-stride over tiles
  mlp_fused_wmma_kernel<<<grid, block, 0, stream>>>(x, W0, b0, W1, b1, W2, b2,
                                                    out, numTiles);
}